// SAM_15264313770034
// MI455X (gfx1250) — compile-verified
//
#include <hip/hip_runtime.h>
#include <math.h>

// ---------------------------------------------------------------------------
// MI455X (gfx1250) implementation of the SAM reference.
// fp32 end-to-end: workload is HBM/scan-latency bound (~95 GFLOP vs ~2 GB
// traffic at 23.3 TB/s), so V_WMMA_F32_16X16X4_F32 gives ample matrix
// throughput without a precision change. wave32 throughout.
// ---------------------------------------------------------------------------

typedef __attribute__((ext_vector_type(2))) float v2f;
typedef __attribute__((ext_vector_type(8))) float v8f;

#define Hdim 256
#define Wdim 256
#define HW   65536     // Hdim*Wdim
#define Bn   8

// ---------------------------------------------------------------------------
// 3x3 conv, pad=1, implicit GEMM with V_WMMA_F32_16X16X4_F32.
// One wave: 16 output pixels (row segment) x (16*NTILES) output channels.
// K = Cin*9 ordered (r,s,cin) -> each K-quad = 4 consecutive channels.
// Block = 128 threads = 4 waves covering 64 pixels of one row.
// NTILES=2 amortizes each A fetch over two cout tiles (halves the re-read
// traffic of the 268MB gated tensor, the dominant HBM consumer).
// Border handling: addresses clamped into the plane (always-safe loads),
// padded lanes zeroed by a mask multiply -> no divergent predication.
// Weights staged in LDS with padded stride (Cin*9 is a multiple of 64 ->
// +1 pad avoids a 16-way bank conflict on the B-fragment reads).
// ---------------------------------------------------------------------------
template <int CIN, bool RELU, int NTILES>
__global__ void conv3x3_wmma(const float* __restrict__ x,
                             const float* __restrict__ wgt,
                             float* __restrict__ out, int Cout)
{
    extern __shared__ float wlds[];              // 16*NTILES*(CIN*9+1) floats
    const int lane     = threadIdx.x & 31;
    const int wave     = threadIdx.x >> 5;
    const int coutBase = blockIdx.y * (16 * NTILES);
    const int b        = blockIdx.z;
    const int h        = blockIdx.x >> 2;                       // 4 blocks/row
    const int w0       = ((blockIdx.x & 3) << 6) + (wave << 4); // 16-pixel seg

    // Cooperative, coalesced weight stage (OIHW, contiguous per cout).
    constexpr int nw      = CIN * 9;
    constexpr int wstride = nw + 1;
    for (int i = threadIdx.x; i < 16 * NTILES * nw; i += 128) {
        int n = i / nw, k = i - n * nw;
        wlds[n * wstride + k] = wgt[(size_t)(coutBase + n) * nw + k];
    }
    __syncthreads();

    const int m     = lane & 15;   // pixel (A rows) / cout (B cols) per lane
    const int halfk = lane >> 4;   // 0 -> K{0,1}, 1 -> K{2,3} of each quad
    const float* xbase  = x + (size_t)b * CIN * HW + (size_t)(halfk << 1) * HW;
    const float* wbase0 = wlds + m * wstride + (halfk << 1) * 9;
    const float* wbase1 = wbase0 + 16 * wstride;   // second cout tile
    v8f acc0 = {};
    v8f acc1 = {};

    #pragma unroll
    for (int r = 0; r < 3; ++r) {
        int  hy    = h + r - 1;
        bool rowok = (unsigned)hy < (unsigned)Hdim;
        int  hyc   = rowok ? hy : 0;
        #pragma unroll
        for (int s = 0; s < 3; ++s) {
            int   wx  = w0 + m + s - 1;
            bool  ok  = rowok && ((unsigned)wx < (unsigned)Wdim);
            int   wxc = ((unsigned)wx < (unsigned)Wdim) ? wx : 0;
            float msk = ok ? 1.f : 0.f;                 // border zero-pad
            const float* xp  = xbase + hyc * Wdim + wxc; // always-valid addr
            const float* wp0 = wbase0 + r * 3 + s;
            const float* wp1 = wbase1 + r * 3 + s;
            for (int cin = 0; cin < CIN; cin += 4) {
                float a0 = xp[(size_t)(cin    ) * HW] * msk;
                float a1 = xp[(size_t)(cin + 1) * HW] * msk;
                v2f A = {a0, a1};
                v2f B0 = { wp0[(cin    ) * 9], wp0[(cin + 1) * 9] };
                acc0 = __builtin_amdgcn_wmma_f32_16x16x4_f32(
                           false, A, false, B0, (short)0, acc0, false, false);
                if (NTILES == 2) {
                    v2f B1 = { wp1[(cin    ) * 9], wp1[(cin + 1) * 9] };
                    acc1 = __builtin_amdgcn_wmma_f32_16x16x4_f32(
                               false, A, false, B1, (short)0, acc1, false, false);
                }
            }
        }
    }

    // D layout: VGPR v -> M = v + 8*halfk, N = lane&15.
    {
        const int n = coutBase + m;
        float* op = out + ((size_t)b * Cout + n) * HW + h * Wdim + w0 + halfk * 8;
        #pragma unroll
        for (int v = 0; v < 8; ++v) {
            float val = acc0[v];
            if (RELU) val = fmaxf(val, 0.f);
            op[v] = val;
        }
    }
    if (NTILES == 2) {
        const int n = coutBase + 16 + m;
        float* op = out + ((size_t)b * Cout + n) * HW + h * Wdim + w0 + halfk * 8;
        #pragma unroll
        for (int v = 0; v < 8; ++v) {
            float val = acc1[v];
            if (RELU) val = fmaxf(val, 0.f);
            op[v] = val;
        }
    }
}

// ---------------------------------------------------------------------------
// Vertical IRNN (axis=H): thread per (b,c,w) column -> coalesced every step.
// Fuses the attention gate: gated[0:32]=td*w[0], gated[64:96]=dt*w[2].
// grid = B*C blocks of 256 threads (W == 256).
// ---------------------------------------------------------------------------
__global__ void irnn_v(const float* __restrict__ xin,
                       const float* __restrict__ alpha,
                       const float* __restrict__ weight,
                       float* __restrict__ gated)
{
    int bc = blockIdx.x;
    int b = bc >> 5, c = bc & 31;
    int w = threadIdx.x;
    float a = alpha[c];
    const float* xp  = xin    + (size_t)bc * HW + w;
    const float* w0p = weight + ((size_t)b * 4 + 0) * HW + w;
    const float* w2p = weight + ((size_t)b * 4 + 2) * HW + w;
    float* g0 = gated + ((size_t)b * 128 +       c) * HW + w;  // up   (td)
    float* g2 = gated + ((size_t)b * 128 +  64 + c) * HW + w;  // down (dt)

    float hp = 0.f;
    for (int hh = 0; hh < Hdim; ++hh) {
        float v = fmaxf(xp[hh * Wdim] + a * hp, 0.f);
        hp = v;
        g0[hh * Wdim] = v * w0p[hh * Wdim];
    }
    hp = 0.f;
    for (int hh = Hdim - 1; hh >= 0; --hh) {
        float v = fmaxf(xp[hh * Wdim] + a * hp, 0.f);
        hp = v;
        g2[hh * Wdim] = v * w2p[hh * Wdim];
    }
}

// ---------------------------------------------------------------------------
// Horizontal IRNN (axis=W): block owns one (b,c) 256x256 plane; 256x32 chunks
// staged through LDS (stride 33 -> conflict free) so global traffic stays
// coalesced while each thread scans its own row. Gate fused on store:
// gated[32:64]=lr*w[1], gated[96:128]=rl*w[3].
// grid = B*C blocks of 256 threads, dynamic LDS = 256*33*4 bytes.
// ---------------------------------------------------------------------------
__global__ void irnn_h(const float* __restrict__ xin,
                       const float* __restrict__ alpha,
                       const float* __restrict__ weight,
                       float* __restrict__ gated)
{
    extern __shared__ float tile[];              // 256 rows * 33
    int bc = blockIdx.x;
    int b = bc >> 5, c = bc & 31;
    float a = alpha[c];
    const float* xp  = xin    + (size_t)bc * HW;
    const float* w1p = weight + ((size_t)b * 4 + 1) * HW;
    const float* w3p = weight + ((size_t)b * 4 + 3) * HW;
    float* g1 = gated + ((size_t)b * 128 + 32 + c) * HW;   // right (lr)
    float* g3 = gated + ((size_t)b * 128 + 96 + c) * HW;   // left  (rl)
    const int t = threadIdx.x;

    // forward: left -> right
    float hp = 0.f;
    for (int chunk = 0; chunk < Wdim / 32; ++chunk) {
        int wbase = chunk * 32;
        __syncthreads();
        for (int rep = 0; rep < 32; ++rep) {
            int lin = rep * 256 + t;
            int hh = lin >> 5, ww = lin & 31;
            tile[hh * 33 + ww] = xp[hh * Wdim + wbase + ww];
        }
        __syncthreads();
        for (int ww = 0; ww < 32; ++ww) {
            float v = fmaxf(tile[t * 33 + ww] + a * hp, 0.f);
            hp = v;
            tile[t * 33 + ww] = v;
        }
        __syncthreads();
        for (int rep = 0; rep < 32; ++rep) {
            int lin = rep * 256 + t;
            int hh = lin >> 5, ww = lin & 31;
            int idx = hh * Wdim + wbase + ww;
            g1[idx] = tile[hh * 33 + ww] * w1p[idx];
        }
    }

    // backward: right -> left
    hp = 0.f;
    for (int chunk = Wdim / 32 - 1; chunk >= 0; --chunk) {
        int wbase = chunk * 32;
        __syncthreads();
        for (int rep = 0; rep < 32; ++rep) {
            int lin = rep * 256 + t;
            int hh = lin >> 5, ww = lin & 31;
            tile[hh * 33 + ww] = xp[hh * Wdim + wbase + ww];
        }
        __syncthreads();
        for (int ww = 31; ww >= 0; --ww) {
            float v = fmaxf(tile[t * 33 + ww] + a * hp, 0.f);
            hp = v;
            tile[t * 33 + ww] = v;
        }
        __syncthreads();
        for (int rep = 0; rep < 32; ++rep) {
            int lin = rep * 256 + t;
            int hh = lin >> 5, ww = lin & 31;
            int idx = hh * Wdim + wbase + ww;
            g3[idx] = tile[hh * 33 + ww] * w3p[idx];
        }
    }
}

// 1x1 conv 16->4 + bias + sigmoid -> attention weight maps (B,4,H,W).
__global__ void att3_sigmoid(const float* __restrict__ a2,
                             const float* __restrict__ w3,
                             const float* __restrict__ b3,
                             float* __restrict__ weight)
{
    int idx = blockIdx.x * blockDim.x + threadIdx.x;   // B*HW threads
    int b = idx >> 16, p = idx & (HW - 1);
    const float* ap = a2 + (size_t)b * 16 * HW + p;
    float s0 = b3[0], s1 = b3[1], s2 = b3[2], s3 = b3[3];
    #pragma unroll
    for (int cc = 0; cc < 16; ++cc) {
        float v = ap[(size_t)cc * HW];
        s0 += v * w3[cc];
        s1 += v * w3[16 + cc];
        s2 += v * w3[32 + cc];
        s3 += v * w3[48 + cc];
    }
    float* wp = weight + (size_t)b * 4 * HW + p;
    wp[0 * HW] = 1.f / (1.f + __expf(-s0));
    wp[1 * HW] = 1.f / (1.f + __expf(-s1));
    wp[2 * HW] = 1.f / (1.f + __expf(-s2));
    wp[3 * HW] = 1.f / (1.f + __expf(-s3));
}

// 1x1 conv 32->1 + sigmoid -> final mask (B,1,H,W).
__global__ void wout_sigmoid(const float* __restrict__ t32,
                             const float* __restrict__ wv,
                             float* __restrict__ outp)
{
    int idx = blockIdx.x * blockDim.x + threadIdx.x;   // B*HW threads
    int b = idx >> 16, p = idx & (HW - 1);
    const float* tp = t32 + (size_t)b * 32 * HW + p;
    float s = 0.f;
    #pragma unroll
    for (int cc = 0; cc < 32; ++cc) s += tp[(size_t)cc * HW] * wv[cc];
    outp[idx] = 1.f / (1.f + __expf(-s));
}

// ---------------------------------------------------------------------------
extern "C" void kernel_launch(void* const* d_in, const int* in_sizes, int n_in,
                              void* d_out, int out_size, void* d_ws, size_t ws_size,
                              hipStream_t stream)
{
    (void)in_sizes; (void)n_in; (void)out_size; (void)ws_size;

    const float* x      = (const float*)d_in[0];
    const float* alpha1 = (const float*)d_in[1];
    const float* alpha2 = (const float*)d_in[2];
    const float* w_in   = (const float*)d_in[3];
    const float* w2     = (const float*)d_in[4];
    const float* w3     = (const float*)d_in[5];
    const float* att_w1 = (const float*)d_in[6];
    const float* att_w2 = (const float*)d_in[7];
    const float* att_w3 = (const float*)d_in[8];
    const float* att_b3 = (const float*)d_in[9];
    const float* w_out  = (const float*)d_in[10];
    float* outp = (float*)d_out;

    // Workspace layout (floats): weight maps | 32ch buffer | 128ch gated.
    float* wsf       = (float*)d_ws;
    float* weightbuf = wsf;                         //  8 MB: (B,4,H,W)
    float* buf32     = weightbuf + (size_t)Bn * 4 * HW;   // 64 MB: (B,32,H,W)
    float* gated     = buf32 + (size_t)Bn * 32 * HW;      // 256 MB: (B,128,H,W)
    float* a1        = gated;                       // attention temps alias
    float* a2        = gated + (size_t)Bn * 16 * HW;//   the not-yet-live gated

    const dim3 cblk(128);
    const dim3 cgrid16(Hdim * (Wdim / 64), 1, Bn);  // Cout tile count 1
    const size_t lds_att1 = 16 * (32  * 9 + 1) * sizeof(float);
    const size_t lds_att2 = 16 * (16  * 9 + 1) * sizeof(float);
    const size_t lds_c32  = 32 * (32  * 9 + 1) * sizeof(float);
    const size_t lds_c128 = 32 * (128 * 9 + 1) * sizeof(float);   // ~147.5 KB
    const size_t ldsH     = 256 * 33 * sizeof(float);

    // CDNA5 allows up to 320 KB LDS per workgroup; lift HIP's 64 KB default
    // for the dynamic-LDS conv kernels (host-side, capture-safe, idempotent).
    hipFuncSetAttribute((const void*)&conv3x3_wmma<32,  false, 2>,
                        hipFuncAttributeMaxDynamicSharedMemorySize, 320 * 1024);
    hipFuncSetAttribute((const void*)&conv3x3_wmma<128, false, 2>,
                        hipFuncAttributeMaxDynamicSharedMemorySize, 320 * 1024);
    hipFuncSetAttribute((const void*)&conv3x3_wmma<128, true,  2>,
                        hipFuncAttributeMaxDynamicSharedMemorySize, 320 * 1024);

    // Attention branch -> per-direction weight maps.
    conv3x3_wmma<32, true, 1><<<cgrid16, cblk, lds_att1, stream>>>(x,  att_w1, a1, 16);
    conv3x3_wmma<16, true, 1><<<cgrid16, cblk, lds_att2, stream>>>(a1, att_w2, a2, 16);
    att3_sigmoid<<<(Bn * HW) / 256, 256, 0, stream>>>(a2, att_w3, att_b3, weightbuf);

    // Trunk (Cout=32 handled as 2 cout tiles per block -> one A fetch feeds
    // two WMMAs, halving input re-read traffic).
    conv3x3_wmma<32, false, 2><<<cgrid16, cblk, lds_c32, stream>>>(x, w_in, buf32, 32);
    irnn_v<<<Bn * 32, 256, 0,    stream>>>(buf32, alpha1, weightbuf, gated);
    irnn_h<<<Bn * 32, 256, ldsH, stream>>>(buf32, alpha1, weightbuf, gated);
    conv3x3_wmma<128, false, 2><<<cgrid16, cblk, lds_c128, stream>>>(gated, w2, buf32, 32);
    irnn_v<<<Bn * 32, 256, 0,    stream>>>(buf32, alpha2, weightbuf, gated);
    irnn_h<<<Bn * 32, 256, ldsH, stream>>>(buf32, alpha2, weightbuf, gated);
    conv3x3_wmma<128, true, 2><<<cgrid16, cblk, lds_c128, stream>>>(gated, w3, buf32, 32);
    wout_sigmoid<<<(Bn * HW) / 256, 256, 0, stream>>>(buf32, w_out, outp);
}